// TensorProductEncoder_86852828659847
// MI455X (gfx1250) — compile-verified
//
#include <hip/hip_runtime.h>

typedef float v2f __attribute__((ext_vector_type(2)));
typedef float v8f __attribute__((ext_vector_type(8)));

#define FD 128      // FILLER_DIM
#define RD 64       // ROLE_DIM
#define FW 768      // FINAL_WIDTH
#define BB 512      // batch
#define SS 512      // seq

// ---- gfx1250 async global->LDS copy (ASYNCcnt path), with safe fallback ----
#if defined(__AMDGCN__) && __has_builtin(__builtin_amdgcn_global_load_async_to_lds_b128)
#define HAVE_ASYNC_LDS 1
typedef int v4i __attribute__((vector_size(16)));
typedef __attribute__((address_space(1))) v4i g_v4i;   // global (AS1)
typedef __attribute__((address_space(3))) v4i l_v4i;   // LDS (AS3)
__device__ __forceinline__ void cp16_async(const void* g, void* l) {
  __builtin_amdgcn_global_load_async_to_lds_b128((g_v4i*)g, (l_v4i*)l, 0, 0);
}
__device__ __forceinline__ void cp_wait() {
#if __has_builtin(__builtin_amdgcn_s_wait_asynccnt)
  __builtin_amdgcn_s_wait_asynccnt(0);
#else
  asm volatile("s_wait_asynccnt 0" ::: "memory");
#endif
}
#else
#define HAVE_ASYNC_LDS 0
#endif

__device__ __forceinline__ void stage16(const float* g, float* l) {
#if HAVE_ASYNC_LDS
  cp16_async(g, l);
#else
  *(float4*)l = *(const float4*)g;
#endif
}

// ---------------------------------------------------------------------------
// Stage 1: per-batch C_b = fe_b^T @ re_b   ([128 x 64], K = 512)
// One 8-wave block per batch; gathered embedding rows streamed into LDS per
// 64-row K chunk via global_load_async_to_lds_b128 (per-lane gather).
// ---------------------------------------------------------------------------
__global__ __launch_bounds__(256) void tpe_outer_kernel(
    const int* __restrict__ fillers, const int* __restrict__ roles,
    const float* __restrict__ filler_table, const float* __restrict__ role_table,
    float* __restrict__ flat_out)
{
  __shared__ float sFe[64 * FD];   // 32 KB, [k][f]
  __shared__ float sRe[64 * RD];   // 16 KB, [k][r]

  const int b    = blockIdx.x;
  const int t    = threadIdx.x;
  const int wave = t >> 5;
  const int lane = t & 31;
  const int hi   = lane >> 4;      // half-wave: selects K pair {0,1} vs {2,3}
  const int l    = lane & 15;
  const int m0   = wave * 16;      // 8 waves cover M = 128

  v8f acc[4] = {};                 // N tiles at 0,16,32,48

  for (int k0 = 0; k0 < SS; k0 += 64) {
    __syncthreads();               // previous chunk fully consumed
    // gather 64 filler rows (128 f32 each) -> LDS, 16B per lane-op
    #pragma unroll
    for (int i = 0; i < 8; ++i) {
      int idx  = t + i * 256;               // 2048 float4 slots
      int row  = idx >> 5;                  // 32 float4 per row
      int c4   = idx & 31;
      int frow = fillers[b * SS + k0 + row];
      stage16(filler_table + (size_t)frow * FD + c4 * 4, &sFe[row * FD + c4 * 4]);
    }
    // gather 64 role rows (64 f32 each) -> LDS
    #pragma unroll
    for (int i = 0; i < 4; ++i) {
      int idx  = t + i * 256;               // 1024 float4 slots
      int row  = idx >> 4;                  // 16 float4 per row
      int c4   = idx & 15;
      int rrow = roles[b * SS + k0 + row];
      stage16(role_table + (size_t)rrow * RD + c4 * 4, &sRe[row * RD + c4 * 4]);
    }
#if HAVE_ASYNC_LDS
    cp_wait();                     // ASYNCcnt==0 before signaling the barrier
#endif
    __syncthreads();

    #pragma unroll
    for (int k = 0; k < 64; k += 4) {
      // A tile 16x4 = fe^T[m0:m0+16, k:k+4]  (fe stored [k][f] in LDS)
      v2f a;
      a.x = sFe[(k + 2 * hi + 0) * FD + m0 + l];
      a.y = sFe[(k + 2 * hi + 1) * FD + m0 + l];
      #pragma unroll
      for (int nt = 0; nt < 4; ++nt) {
        v2f bf;
        bf.x = sRe[(k + 2 * hi + 0) * RD + nt * 16 + l];
        bf.y = sRe[(k + 2 * hi + 1) * RD + nt * 16 + l];
        acc[nt] = __builtin_amdgcn_wmma_f32_16x16x4_f32(
            false, a, false, bf, (short)0, acc[nt], false, false);
      }
    }
  }

  // C/D layout: VGPR r -> M = r (lanes 0-15) or r+8 (lanes 16-31), N = lane%16
  float* out_b = flat_out + (size_t)b * (FD * RD);
  #pragma unroll
  for (int nt = 0; nt < 4; ++nt)
    #pragma unroll
    for (int r = 0; r < 8; ++r) {
      int M = m0 + r + 8 * hi;
      int N = nt * 16 + l;
      out_b[M * RD + N] = acc[nt][r];
    }
}

// ---------------------------------------------------------------------------
// Stage 2: out[512,768] = flat[512,8192] @ W^T[8192,768] + bias
// 32x64 output tile per block, 8 waves x one 16x16 accumulator, K chunk 32.
// Both operands kept row-major [row][k] with stride 40 (16B-aligned, avoids
// the 8-way bank conflicts a stride-32 layout would give on fragment reads).
// ---------------------------------------------------------------------------
#define SA_STR 40
#define SW_STR 40

__global__ __launch_bounds__(256) void tpe_final_kernel(
    const float* __restrict__ flat, const float* __restrict__ W,
    const float* __restrict__ bias, float* __restrict__ out)
{
  __shared__ float sA[32 * SA_STR];   // 5 KB,  [m][k]
  __shared__ float sW[64 * SW_STR];   // 10 KB, [n][k]

  const int K    = FD * RD;       // 8192
  const int t    = threadIdx.x;
  const int wave = t >> 5;
  const int lane = t & 31;
  const int hi   = lane >> 4;
  const int l    = lane & 15;
  const int m0   = blockIdx.x * 32;    // 16 blocks over M = 512
  const int n0   = blockIdx.y * 64;    // 12 blocks over N = 768
  const int mw   = (wave >> 2) * 16;   // 0 | 16
  const int nw   = (wave & 3) * 16;    // 0 | 16 | 32 | 48

  v8f acc = {};
  const float bn = bias[n0 + nw + l];  // K-invariant epilogue operand

  for (int k0 = 0; k0 < K; k0 += 32) {
    __syncthreads();
    {   // sA: 32 rows x 32 k = 256 float4 copies, one per thread
      int row = t >> 3;
      int c4  = t & 7;
      stage16(flat + (size_t)(m0 + row) * K + k0 + c4 * 4,
              &sA[row * SA_STR + c4 * 4]);
    }
    // sW: 64 rows x 32 k = 512 float4 copies, two per thread (no transpose)
    #pragma unroll
    for (int i = 0; i < 2; ++i) {
      int idx = t + i * 256;
      int row = idx >> 3;
      int c4  = idx & 7;
      stage16(W + (size_t)(n0 + row) * K + k0 + c4 * 4,
              &sW[row * SW_STR + c4 * 4]);
    }
#if HAVE_ASYNC_LDS
    cp_wait();
#endif
    __syncthreads();

    #pragma unroll
    for (int k = 0; k < 32; k += 4) {
      v2f a;   // A[m][k..k+1 | k+2..k+3]: consecutive -> ds_load_b64
      a.x = sA[(mw + l) * SA_STR + k + 2 * hi + 0];
      a.y = sA[(mw + l) * SA_STR + k + 2 * hi + 1];
      v2f bf;  // B[k][n] = W[n][k]: lane = N, half-wave = K pair
      bf.x = sW[(nw + l) * SW_STR + k + 2 * hi + 0];
      bf.y = sW[(nw + l) * SW_STR + k + 2 * hi + 1];
      acc = __builtin_amdgcn_wmma_f32_16x16x4_f32(
          false, a, false, bf, (short)0, acc, false, false);
    }
  }

  #pragma unroll
  for (int r = 0; r < 8; ++r) {
    int M = m0 + mw + r + 8 * hi;
    int N = n0 + nw + l;
    out[(size_t)M * FW + N] = acc[r] + bn;
  }
}

// ---------------------------------------------------------------------------
extern "C" void kernel_launch(void* const* d_in, const int* in_sizes, int n_in,
                              void* d_out, int out_size, void* d_ws, size_t ws_size,
                              hipStream_t stream) {
  const int*   fillers      = (const int*)d_in[0];
  const int*   roles        = (const int*)d_in[1];
  const float* filler_table = (const float*)d_in[2];
  const float* role_table   = (const float*)d_in[3];
  const float* W_last       = (const float*)d_in[4];
  const float* b_last       = (const float*)d_in[5];
  float* out  = (float*)d_out;
  float* flat = (float*)d_ws;   // [512, 8192] fp32 intermediate = 16.8 MB

  tpe_outer_kernel<<<BB, 256, 0, stream>>>(fillers, roles, filler_table,
                                           role_table, flat);
  dim3 g2(BB / 32, FW / 64);
  tpe_final_kernel<<<g2, 256, 0, stream>>>(flat, W_last, b_last, out);
}